// PQCClassifier_5119601017550
// MI455X (gfx1250) — compile-verified
//
#include <hip/hip_runtime.h>
#include <hip/hip_bf16.h>

// PQC classifier: output is independent of x (state starts at |0000>, gates are
// parameterized only by the three scalar thetas). Whole problem reduces to:
//   logits = [cos(tx) + cos(ty), 2]  ->  log_softmax  ->  broadcast to 1M rows.
// 8 MiB of stores @ 23.3 TB/s ~ 0.36us; we never touch the 64 MiB x tensor.
// CDNA5-specific path exercised: async LDS->global store DMA + s_wait_asynccnt.

#define AS1 __attribute__((address_space(1)))
#define AS3 __attribute__((address_space(3)))

// Pointee type required by the async-store builtin (per hipcc diagnostic):
// pointer to a 16-byte int vector in AS1 (global) / AS3 (LDS).
typedef int v4i __attribute__((vector_size(4 * sizeof(int))));

#if defined(__has_builtin)
#  if __has_builtin(__builtin_amdgcn_global_store_async_from_lds_b128)
#    define HAVE_ASYNC_STORE 1
#  endif
#  if __has_builtin(__builtin_amdgcn_s_wait_asynccnt)
#    define HAVE_WAIT_ASYNC 1
#  endif
#endif

// ---------------------------------------------------------------------------
// Kernel 1: compute the two log-softmax values once (single lane of work).
// Mirrors the reference numerics: half-angle cos/sin in f32, |amp|^2 via
// squared magnitudes, including the ~1.0 normalization factors so float
// rounding tracks the JAX reference to ~1e-7.
// ---------------------------------------------------------------------------
__global__ void pqc_logits_kernel(const float* __restrict__ theta_rx,
                                  const float* __restrict__ theta_ry,
                                  const float* __restrict__ theta_rz,
                                  float* __restrict__ ws) {
  if (threadIdx.x == 0) {
    const float hx = 0.5f * theta_rx[0];
    const float hy = 0.5f * theta_ry[0];
    const float hz = 0.5f * theta_rz[0];
    const float cx = cosf(hx), sx = sinf(hx);
    const float cy = cosf(hy), sy = sinf(hy);
    const float cz = cosf(hz), sz = sinf(hz);
    const float nx = cx * cx + sx * sx;   // ~1
    const float ny = cy * cy + sy * sy;   // ~1
    const float nz = cz * cz + sz * sz;   // ~1 (RZ phase has unit magnitude)
    // <Z> per wire on the product state
    const float m0 = (cx * cx - sx * sx) * ny * nz;  // cos(tx)
    const float m1 = nx * (cy * cy - sy * sy) * nz;  // cos(ty)
    const float m2 = nx * ny * nz;                   // 1 (wire 2)
    const float m3 = m2;                             // 1 (wire 3, untouched)
    const float a = m0 + m1;   // logit 0
    const float b = m2 + m3;   // logit 1
    // log_softmax (max-shifted, like jax.nn.log_softmax)
    const float mx  = fmaxf(a, b);
    const float lse = mx + logf(expf(a - mx) + expf(b - mx));
    ws[0] = a - lse;
    ws[1] = b - lse;
  }
}

// ---------------------------------------------------------------------------
// Kernel 2: broadcast {l0,l1} to all rows. Each thread covers one float4
// (= 2 output rows). Preferred path: stage the pattern in LDS (replicated
// 16 B per lane, 16 B aligned) and fire the CDNA5 async LDS->global DMA.
// Fallback: plain 128-bit vector store (still saturates store bandwidth).
// ---------------------------------------------------------------------------
__global__ void pqc_bcast_kernel(const float* __restrict__ ws,
                                 float* __restrict__ out,
                                 int nquads) {
  // Uniform address -> compiler emits a scalar load; broadcast to all lanes.
  const float l0 = ws[0];
  const float l1 = ws[1];
  const int idx = blockIdx.x * blockDim.x + threadIdx.x;

#if defined(HAVE_ASYNC_STORE)
  __shared__ __align__(16) float lds_pat[128];   // 512 B = 16 B per lane
  const int t = threadIdx.x;
  if (t < 128) lds_pat[t] = (t & 1) ? l1 : l0;   // {l0,l1,l0,l1,...}
  __syncthreads();                                // DS writes visible to DMA

  if (idx < nquads) {
    float* gdst = out + (size_t)idx * 4;
    // generic global address == AS1 address on amdgcn
    AS1 v4i* g = (AS1 v4i*)(unsigned long long)gdst;
    // low 32 bits of a generic LDS address are the LDS byte offset (AS3 is 32-bit)
    AS3 v4i* l = (AS3 v4i*)(unsigned int)(unsigned long long)
                 (&lds_pat[(t & 31) * 4]);
    __builtin_amdgcn_global_store_async_from_lds_b128(g, l, /*offset=*/0,
                                                      /*cpol=*/0);
  }
#  if defined(HAVE_WAIT_ASYNC)
  __builtin_amdgcn_s_wait_asynccnt(0);
#  else
  asm volatile("s_wait_asynccnt 0" ::: "memory");
#  endif
#else
  if (idx < nquads) {
    float4 v = make_float4(l0, l1, l0, l1);
    reinterpret_cast<float4*>(out)[idx] = v;    // global_store_b128
  }
#endif
}

// ---------------------------------------------------------------------------
// Launch: d_in = { x (unused), theta_rx, theta_ry, theta_rz }, out = (bsz, 2).
// ---------------------------------------------------------------------------
extern "C" void kernel_launch(void* const* d_in, const int* in_sizes, int n_in,
                              void* d_out, int out_size, void* d_ws, size_t ws_size,
                              hipStream_t stream) {
  const float* theta_rx = (const float*)d_in[1];
  const float* theta_ry = (const float*)d_in[2];
  const float* theta_rz = (const float*)d_in[3];
  float* out = (float*)d_out;
  float* ws  = (float*)d_ws;

  pqc_logits_kernel<<<1, 32, 0, stream>>>(theta_rx, theta_ry, theta_rz, ws);

  // out_size = 2 * bsz = 2,097,152 floats; always a multiple of 4 here.
  const int nquads  = out_size >> 2;
  const int threads = 256;                       // 8 wave32s per workgroup
  const int blocks  = (nquads + threads - 1) / threads;
  pqc_bcast_kernel<<<blocks, threads, 0, stream>>>(ws, out, nquads);
}